// MultiHeadAttention_16939351015590
// MI455X (gfx1250) — compile-verified
//
#include <hip/hip_runtime.h>
#include <float.h>

#define B_  4
#define P_  2048
#define E_  1024
#define H_  16
#define HD_ 64

typedef __attribute__((ext_vector_type(2))) float v2f;
typedef __attribute__((ext_vector_type(8))) float v8f;

// ---------------------------------------------------------------------------
// Stage 1: Qs[b,h,p] = sum_e Q[b,p,h*64+e]; same for Ks.  (n indexes h fastest
// so a wave touches contiguous rows)
// ---------------------------------------------------------------------------
__global__ void rowsum_k(const float* __restrict__ Q, const float* __restrict__ K,
                         float* __restrict__ Qs, float* __restrict__ Ks) {
    int n = blockIdx.x * blockDim.x + threadIdx.x;   // [0, 2*B*H*P)
    const float* src; float* dst;
    int m = n;
    if (m >= B_ * H_ * P_) { m -= B_ * H_ * P_; src = K; dst = Ks; }
    else                   { src = Q; dst = Qs; }
    int h  = m % H_;
    int bp = m / H_;            // b*P + p
    int p  = bp % P_;
    int b  = bp / P_;
    const float4* ptr = (const float4*)(src + (size_t)bp * E_ + h * HD_);
    float acc = 0.f;
#pragma unroll
    for (int u = 0; u < HD_ / 4; ++u) {
        float4 v = ptr[u];
        acc += v.x + v.y + v.z + v.w;
    }
    dst[((size_t)b * H_ + h) * P_ + p] = acc;
}

// ---------------------------------------------------------------------------
// Stage 2a: partial column sums of V over p-tiles of 256 rows (coalesced).
// Vp[b][ptile][col], ptile in [0,8)
// ---------------------------------------------------------------------------
__global__ void colsum_part_k(const float* __restrict__ V, float* __restrict__ Vp) {
    int t     = threadIdx.x;
    int ctile = blockIdx.x % (E_ / 256);
    int ptile = (blockIdx.x / (E_ / 256)) % 8;
    int b     = blockIdx.x / ((E_ / 256) * 8);
    int col   = ctile * 256 + t;
    const float* base = V + ((size_t)b * P_ + (size_t)ptile * 256) * E_ + col;
    float acc = 0.f;
    for (int r = 0; r < 256; ++r) acc += base[(size_t)r * E_];
    Vp[((size_t)b * 8 + ptile) * E_ + col] = acc;
}

// Stage 2b: Vs[b][col] = sum over 8 partials
__global__ void colsum_fin_k(const float* __restrict__ Vp, float* __restrict__ Vs) {
    int n   = blockIdx.x * blockDim.x + threadIdx.x;  // [0, B*E)
    int col = n % E_;
    int b   = n / E_;
    float acc = 0.f;
#pragma unroll
    for (int pt = 0; pt < 8; ++pt) acc += Vp[((size_t)b * 8 + pt) * E_ + col];
    Vs[n] = acc;
}

// ---------------------------------------------------------------------------
// Stage 3: diagT[b][p][h] = softmax_k(Qs_i*Ks_k*s)[i==k].  One block per (b,h).
// Numerically safe: m_i = a_i>=0 ? a_i*maxK : a_i*minK  (exact row max).
// ---------------------------------------------------------------------------
__global__ void diag_k(const float* __restrict__ Qs, const float* __restrict__ Ks,
                       float* __restrict__ diagT) {
    __shared__ float ks[P_];
    __shared__ float red[2][256];
    int bh = blockIdx.x;          // b*H + h
    int t  = threadIdx.x;
    int h  = bh % H_;
    int b  = bh / H_;
    const float* ksrc = Ks + (size_t)bh * P_;
    float lmax = -FLT_MAX, lmin = FLT_MAX;
    for (int i = t; i < P_; i += 256) {
        float v = ksrc[i];
        ks[i] = v;
        lmax = fmaxf(lmax, v);
        lmin = fminf(lmin, v);
    }
    red[0][t] = lmax; red[1][t] = lmin;
    __syncthreads();
    for (int s = 128; s > 0; s >>= 1) {
        if (t < s) {
            red[0][t] = fmaxf(red[0][t], red[0][t + s]);
            red[1][t] = fminf(red[1][t], red[1][t + s]);
        }
        __syncthreads();
    }
    float kmax = red[0][0], kmin = red[1][0];

    const float* qsrc = Qs + (size_t)bh * P_;
    const float  sc   = 0.125f;   // 1/sqrt(64)
    float a[8], m[8], denom[8];
#pragma unroll
    for (int u = 0; u < 8; ++u) {
        float av = qsrc[t + 256 * u] * sc;
        a[u]     = av;
        m[u]     = (av >= 0.f) ? av * kmax : av * kmin;
        denom[u] = 0.f;
    }
    for (int k = 0; k < P_; ++k) {
        float kv = ks[k];
#pragma unroll
        for (int u = 0; u < 8; ++u) denom[u] += __expf(a[u] * kv - m[u]);
    }
#pragma unroll
    for (int u = 0; u < 8; ++u) {
        int i = t + 256 * u;
        float num = __expf(a[u] * ks[i] - m[u]);
        diagT[((size_t)b * P_ + i) * H_ + h] = num / denom[u];
    }
}

// ---------------------------------------------------------------------------
// Stage 4: G[b][h][j] = sum_e Vs[b][h*64+e] * W[j][h*64+e].
// One block per (b, 16-row W tile); W tile staged in LDS (coalesced).
// ---------------------------------------------------------------------------
__global__ void g_k(const float* __restrict__ Vs, const float* __restrict__ W,
                    float* __restrict__ G) {
    __shared__ float wl[16][E_ + 1];
    int t  = threadIdx.x;
    int jt = blockIdx.x % (E_ / 16);
    int b  = blockIdx.x / (E_ / 16);
    int j0 = jt * 16;
    for (int idx = t; idx < 16 * E_; idx += 256) {
        int r = idx / E_, c = idx % E_;
        wl[r][c] = W[(size_t)(j0 + r) * E_ + c];
    }
    __syncthreads();
    int h  = t / 16;
    int jl = t % 16;
    const float* vsrc = Vs + (size_t)b * E_ + h * HD_;
    float acc = 0.f;
#pragma unroll
    for (int e = 0; e < HD_; ++e) acc += vsrc[e] * wl[jl][h * HD_ + e];
    G[((size_t)b * H_ + h) * E_ + j0 + jl] = acc;
}

// ---------------------------------------------------------------------------
// Stage 5: OUT[b,p,j] = sum_h diagT[b][p][h] * G[b][h][j] + bias[j]
// fp32 WMMA: four V_WMMA_F32_16X16X4_F32 cover K=16.
// Block = 8 waves; one M-tile (16 rows of p) per block; each wave does 8
// N-tiles reusing its A fragments.  EXEC is uniformly all-ones.
// ---------------------------------------------------------------------------
__global__ void out_wmma_k(const float* __restrict__ diagT, const float* __restrict__ G,
                           const float* __restrict__ bias, float* __restrict__ OUT) {
    int t    = threadIdx.x;
    int lane = t & 31;
    int wave = t >> 5;
    int mt   = blockIdx.x % (P_ / 16);
    int b    = blockIdx.x / (P_ / 16);
    int m0   = mt * 16;
    int hl   = lane & 15;   // row (A) / col (B,C,D) index within tile
    int hi   = lane >> 4;   // lane-half selects K pairs {0,1} vs {2,3}

    // A fragment slices: A[M=p, K=h] = diagT[b][p][h]; K pair per VGPR slot.
    const float* arow = diagT + ((size_t)b * P_ + (m0 + hl)) * H_;
    v2f afrag[4];
#pragma unroll
    for (int s = 0; s < 4; ++s) {
        int k0 = 4 * s + 2 * hi;
        afrag[s].x = arow[k0];
        afrag[s].y = arow[k0 + 1];
    }

    const float* gb   = G + (size_t)b * H_ * E_;
    float*       outb = OUT + (size_t)b * P_ * E_;

    for (int nt = wave; nt < E_ / 16; nt += 8) {   // uniform trip count
        int n0 = nt * 16;
        float bv = bias[n0 + hl];
        v8f c;
#pragma unroll
        for (int r = 0; r < 8; ++r) c[r] = bv;     // bias constant per column
#pragma unroll
        for (int s = 0; s < 4; ++s) {
            int k0 = 4 * s + 2 * hi;
            v2f bfrag;
            bfrag.x = gb[(size_t)k0 * E_ + n0 + hl];
            bfrag.y = gb[(size_t)(k0 + 1) * E_ + n0 + hl];
            c = __builtin_amdgcn_wmma_f32_16x16x4_f32(
                    false, afrag[s], false, bfrag, (short)0, c, false, false);
        }
        // D layout: VGPR r -> (M=m0+r+8*hi, N=n0+hl)
#pragma unroll
        for (int r = 0; r < 8; ++r) {
            int m = m0 + r + 8 * hi;
            outb[(size_t)m * E_ + n0 + hl] = c[r];
        }
    }
}

// ---------------------------------------------------------------------------
extern "C" void kernel_launch(void* const* d_in, const int* in_sizes, int n_in,
                              void* d_out, int out_size, void* d_ws, size_t ws_size,
                              hipStream_t stream) {
    const float* Q    = (const float*)d_in[0];
    const float* K    = (const float*)d_in[1];
    const float* V    = (const float*)d_in[2];
    const float* W    = (const float*)d_in[3];
    const float* bias = (const float*)d_in[4];
    float* out = (float*)d_out;
    float* ws  = (float*)d_ws;

    float* Qs = ws;                 // B*H*P          = 131072
    float* Ks = ws + 131072;        // B*H*P          = 131072
    float* Vp = ws + 262144;        // B*8*E          =  32768
    float* Vs = ws + 294912;        // B*E            =   4096
    float* dT = ws + 299008;        // B*P*H          = 131072
    float* G  = ws + 430080;        // B*H*E          =  65536
    // total 495616 floats (~1.9 MB) of workspace

    rowsum_k    <<<(2 * B_ * H_ * P_) / 256, 256, 0, stream>>>(Q, K, Qs, Ks);
    colsum_part_k<<<B_ * 8 * (E_ / 256), 256, 0, stream>>>(V, Vp);
    colsum_fin_k<<<(B_ * E_) / 256, 256, 0, stream>>>(Vp, Vs);
    diag_k      <<<B_ * H_, 256, 0, stream>>>(Qs, Ks, dT);
    g_k         <<<B_ * (E_ / 16), 256, 0, stream>>>(Vs, W, G);
    out_wmma_k  <<<B_ * (P_ / 16), 256, 0, stream>>>(dT, G, bias, out);
}